// WeightedGCN4_27504970564050
// MI455X (gfx1250) — compile-verified
//
#include <hip/hip_runtime.h>
#include <hip/hip_bf16.h>

#define HID    128
#define NFEATC 14000
#define NCELLC 10000
#define NOUTC  134
#define NOUTP  144     // NOUTC padded to multiple of 16
#define NLAYER 2
#define LIN_WAVES 4

typedef float v2f __attribute__((ext_vector_type(2)));
typedef float v8f __attribute__((ext_vector_type(8)));

__device__ __forceinline__ float gelu_f(float x) {
    // jax.nn.gelu default: tanh approximation
    const float c0 = 0.7978845608028654f;
    return 0.5f * x * (1.0f + tanhf(c0 * (x + 0.044715f * x * x * x)));
}
__device__ __forceinline__ float lrelu_f(float x) { return x > 0.0f ? x : 0.01f * x; }

__device__ __forceinline__ float wave_sum(float v) {
#pragma unroll
    for (int off = 16; off > 0; off >>= 1) v += __shfl_xor(v, off, 32);
    return v;
}

// ---------------------------------------------------------------------------
// Y[r,n] = act( sum_k X[r,k]W[k,n] (+ sum_k X2[r,k]W2[k,n]) + bias[n] )
// fp32 WMMA 16x16x4; one wave = one 16x16 output tile.
// REQUIRES: R % 16 == 0, K % 4 == 0, and every launched tile fully in-bounds
// (pad W/Y buffers for ragged N). No guards -> clean unconditional loads.
// A frag: lane m=l&15, VGPR0/1 = K kb+(l>>4)*2+{0,1}  (one float2 load)
// B frag: lane n=l&15, VGPR0/1 = same K rows at column n (two strided loads)
// C/D   : lane n=l&15, VGPR r = row (l>>4)*8+r
// ---------------------------------------------------------------------------
__global__ void linear_wmma_kernel(const float* __restrict__ X,
                                   const float* __restrict__ W,
                                   const float* __restrict__ X2,
                                   const float* __restrict__ W2,
                                   const float* __restrict__ bias,
                                   float* __restrict__ Y,
                                   int R, int K, int N, int act_gelu)
{
    const int wave = threadIdx.x >> 5;
    const int lane = threadIdx.x & 31;
    const int row0 = blockIdx.x << 4;
    const int col0 = (blockIdx.y * LIN_WAVES + wave) << 4;
    if (col0 >= N) return;                       // wave-uniform exit

    const int mn   = lane & 15;                  // M for A, N for B/C
    const int ksel = (lane >> 4) << 1;           // 0 or 2
    const float* __restrict__ xr = X + (size_t)(row0 + mn) * K + ksel;
    const float* __restrict__ wc = W + (col0 + mn) + (size_t)ksel * N;

    v8f acc = {};
#pragma unroll 4
    for (int kb = 0; kb < K; kb += 4) {
        v2f a = *(const v2f*)(xr + kb);          // 8B aligned (ksel even)
        v2f b;
        b[0] = wc[(size_t)kb * N];
        b[1] = wc[(size_t)kb * N + N];
        acc = __builtin_amdgcn_wmma_f32_16x16x4_f32(false, a, false, b,
                                                    (short)0, acc, false, false);
    }
    if (X2 != nullptr) {
        const float* __restrict__ xr2 = X2 + (size_t)(row0 + mn) * K + ksel;
        const float* __restrict__ wc2 = W2 + (col0 + mn) + (size_t)ksel * N;
#pragma unroll 4
        for (int kb = 0; kb < K; kb += 4) {
            v2f a = *(const v2f*)(xr2 + kb);
            v2f b;
            b[0] = wc2[(size_t)kb * N];
            b[1] = wc2[(size_t)kb * N + N];
            acc = __builtin_amdgcn_wmma_f32_16x16x4_f32(false, a, false, b,
                                                        (short)0, acc, false, false);
        }
    }
    const int col = col0 + mn;
    const float bv = bias[col];
    const int mbase = (lane >> 4) << 3;
    float* __restrict__ yp = Y + (size_t)(row0 + mbase) * N + col;
#pragma unroll
    for (int r = 0; r < 8; ++r) {
        float v = acc[r] + bv;
        if (act_gelu) v = gelu_f(v);
        yp[(size_t)r * N] = v;
    }
}

// ---------------------------------------------------------------------------
__global__ void fill_kernel(float* __restrict__ p, float v, size_t n) {
    size_t i = (size_t)blockIdx.x * blockDim.x + threadIdx.x;
    if (i < n) p[i] = v;
}

// zero-pad readout1 weights/bias: Wp[256x144] <- W[256x134], bp[144] <- b[134]
__global__ void pad_w_kernel(const float* __restrict__ W, const float* __restrict__ b,
                             float* __restrict__ Wp, float* __restrict__ bp)
{
    int i = blockIdx.x * blockDim.x + threadIdx.x;    // over 256*144
    if (i < 2 * HID * NOUTP) {
        int k = i / NOUTP, n = i - k * NOUTP;
        Wp[i] = (n < NOUTC) ? W[k * NOUTC + n] : 0.0f;
        if (k == 0) bp[n] = (n < NOUTC) ? b[n] : 0.0f;
    }
}

// d_out[r, 0:134] = Yp[r, 0:134]
__global__ void slice_out_kernel(const float* __restrict__ Yp, float* __restrict__ out)
{
    int i = blockIdx.x * blockDim.x + threadIdx.x;    // over NCELLC*NOUTC
    if (i < NCELLC * NOUTC) {
        int r = i / NOUTC, n = i - r * NOUTC;
        out[i] = Yp[(size_t)r * NOUTP + n];
    }
}

// out[r,:] = lrelu(table[ids[r],:]) — wave per row, float4 per lane
__global__ void embed_lrelu_kernel(const float* __restrict__ table,
                                   const int* __restrict__ ids,
                                   float* __restrict__ out, int R)
{
    int r = blockIdx.x * (blockDim.x >> 5) + (threadIdx.x >> 5);
    if (r >= R) return;
    int lane = threadIdx.x & 31;
    int id = ids[r];
    float4 v = ((const float4*)(table + (size_t)id * HID))[lane];
    v.x = lrelu_f(v.x); v.y = lrelu_f(v.y); v.z = lrelu_f(v.z); v.w = lrelu_f(v.w);
    ((float4*)(out + (size_t)r * HID))[lane] = v;
}

// segment-sum: acc[dst[e],:] += w[e]*X[src[e],:] ; cnt[dst[e]] += 1  (wave per edge)
__global__ void edge_scatter_kernel(const float* __restrict__ X,
                                    const int* __restrict__ src,
                                    const int* __restrict__ dst,
                                    const float* __restrict__ w,
                                    float* __restrict__ acc,
                                    float* __restrict__ cnt, int E)
{
    int e = blockIdx.x * (blockDim.x >> 5) + (threadIdx.x >> 5);
    if (e >= E) return;
    int lane = threadIdx.x & 31;
    int s = src[e], d = dst[e];
    float ww = w[e];
    float4 v = ((const float4*)(X + (size_t)s * HID))[lane];
    float* a = acc + (size_t)d * HID + lane * 4;
    atomicAdd(a + 0, ww * v.x);
    atomicAdd(a + 1, ww * v.y);
    atomicAdd(a + 2, ww * v.z);
    atomicAdd(a + 3, ww * v.w);
    if (lane == 0) atomicAdd(cnt + d, 1.0f);
}

// acc[r,:] /= max(cnt[r],1)
__global__ void seg_div_kernel(float* __restrict__ acc, const float* __restrict__ cnt, int R)
{
    int r = blockIdx.x * (blockDim.x >> 5) + (threadIdx.x >> 5);
    if (r >= R) return;
    int lane = threadIdx.x & 31;
    float inv = 1.0f / fmaxf(cnt[r], 1.0f);
    float4* a = (float4*)(acc + (size_t)r * HID);
    float4 v = a[lane];
    v.x *= inv; v.y *= inv; v.z *= inv; v.w *= inv;
    a[lane] = v;
}

// mode 0: LN(x)   mode 1: LN(gelu(x))   mode 2: gelu(LN(x))
// optionally mirrors result to Y2 (row stride ld2) for the readout concat
__global__ void rownorm_kernel(const float* __restrict__ X, float* __restrict__ Y,
                               const float* __restrict__ g, const float* __restrict__ b,
                               int R, int mode, float* __restrict__ Y2, int ld2)
{
    int r = blockIdx.x * (blockDim.x >> 5) + (threadIdx.x >> 5);
    if (r >= R) return;
    int lane = threadIdx.x & 31;
    float4 v = ((const float4*)(X + (size_t)r * HID))[lane];
    if (mode == 1) { v.x = gelu_f(v.x); v.y = gelu_f(v.y); v.z = gelu_f(v.z); v.w = gelu_f(v.w); }
    float mean = wave_sum(v.x + v.y + v.z + v.w) * (1.0f / HID);
    float dx = v.x - mean, dy = v.y - mean, dz = v.z - mean, dw = v.w - mean;
    float var = wave_sum(dx * dx + dy * dy + dz * dz + dw * dw) * (1.0f / HID);
    float inv = rsqrtf(var + 1e-5f);
    float4 gg = ((const float4*)g)[lane];
    float4 bb = ((const float4*)b)[lane];
    float4 o;
    o.x = dx * inv * gg.x + bb.x;
    o.y = dy * inv * gg.y + bb.y;
    o.z = dz * inv * gg.z + bb.z;
    o.w = dw * inv * gg.w + bb.w;
    if (mode == 2) { o.x = gelu_f(o.x); o.y = gelu_f(o.y); o.z = gelu_f(o.z); o.w = gelu_f(o.w); }
    ((float4*)(Y + (size_t)r * HID))[lane] = o;
    if (Y2) ((float4*)(Y2 + (size_t)r * ld2))[lane] = o;
}

// softmax over {dot(h1,q), dot(h2,q)}; out = gelu(a1*h1 + a2*h2)
__global__ void attention_kernel(const float* __restrict__ h1, const float* __restrict__ h2,
                                 const float* __restrict__ q, float* __restrict__ out, int R)
{
    int r = blockIdx.x * (blockDim.x >> 5) + (threadIdx.x >> 5);
    if (r >= R) return;
    int lane = threadIdx.x & 31;
    float4 a  = ((const float4*)(h1 + (size_t)r * HID))[lane];
    float4 c  = ((const float4*)(h2 + (size_t)r * HID))[lane];
    float4 qq = ((const float4*)(q  + (size_t)r * HID))[lane];
    float s1 = wave_sum(a.x * qq.x + a.y * qq.y + a.z * qq.z + a.w * qq.w);
    float s2 = wave_sum(c.x * qq.x + c.y * qq.y + c.z * qq.z + c.w * qq.w);
    float mx = fmaxf(s1, s2);
    float e1 = __expf(s1 - mx), e2 = __expf(s2 - mx);
    float rs = 1.0f / (e1 + e2);
    float w1 = e1 * rs, w2 = e2 * rs;
    float4 o;
    o.x = gelu_f(w1 * a.x + w2 * c.x);
    o.y = gelu_f(w1 * a.y + w2 * c.y);
    o.z = gelu_f(w1 * a.z + w2 * c.z);
    o.w = gelu_f(w1 * a.w + w2 * c.w);
    ((float4*)(out + (size_t)r * HID))[lane] = o;
}

// ---------------------------------------------------------------------------
extern "C" void kernel_launch(void* const* d_in, const int* in_sizes, int n_in,
                              void* d_out, int out_size, void* d_ws, size_t ws_size,
                              hipStream_t stream)
{
    const int*   feature_ids = (const int*)d_in[0];
    const int*   cell_ids    = (const int*)d_in[1];
    const int*   occur_src   = (const int*)d_in[2];
    const int*   occur_dst   = (const int*)d_in[3];
    const float* occur_w     = (const float*)d_in[4];
    const int*   entail_src  = (const int*)d_in[5];
    const int*   entail_dst  = (const int*)d_in[6];
    const float* entail_w    = (const float*)d_in[7];
    const int*   pathway_src = (const int*)d_in[8];
    const int*   pathway_dst = (const int*)d_in[9];
    const float* pathway_w   = (const float*)d_in[10];
    const int E_OCC = in_sizes[2], E_ENT = in_sizes[5], E_PW = in_sizes[8];
    // params flattened in insertion order starting at index 11:
    // 11 embed_feat, 12 embed_cell, 13/14 in_lin0 W/b, 15/16 in_lin1 W/b,
    // 17/18 in_ln0 g/b, 19/20 in_ln1 g/b,
    // 21+9l conv[l]: entail(Wself,Wneigh,b), occur(...), pathway(...),
    // 39+6l conv_ln[l][j] g/b, 51+2l att[l] W/b, 55..58 readout.
    auto P = [&](int i) { return (const float*)d_in[i]; };

    float* ws = (float*)d_ws;
    size_t off = 0;
    auto alloc = [&](size_t n) { float* p = ws + off; off += n; return p; };
    const size_t NF128 = (size_t)NFEATC * HID;
    const size_t NC128 = (size_t)NCELLC * HID;
    const size_t NC256 = (size_t)NCELLC * HID * 2;
    float* hfeat   = alloc(NF128);
    float* hcell   = alloc(NC128);
    float* acc_ent = alloc(NF128);   // neighbor accum (entail) ; reused as h1
    float* acc_pw  = alloc(NF128);   // neighbor accum (pathway); reused as h2
    float* acc_occ = alloc(NC128);   // neighbor accum (occur)  ; stage1 z_cell
    float* cnt_ent = alloc(NFEATC);
    float* cnt_pw  = alloc(NFEATC);
    float* cnt_occ = alloc(NCELLC);
    float* f_ent   = alloc(NF128);   // stage1 tmp_feat
    float* f_pw    = alloc(NF128);
    float* c_occ   = alloc(NC128);   // stage1 tmp_cell
    float* q       = alloc(NF128);   // stage1 z_feat
    float* hist    = alloc(NC256);
    float* hread   = alloc(NC256);
    float* Wp      = alloc((size_t)2 * HID * NOUTP);   // padded readout1 W
    float* bp      = alloc(NOUTP);                     // padded readout1 b
    float* Yp      = alloc((size_t)NCELLC * NOUTP);    // padded output
    (void)ws_size; (void)n_in; (void)out_size;

    const dim3 TB(32 * LIN_WAVES);                   // 4 waves / block for WMMA kernels
    const int RWB = 256;                             // wave-per-row kernels: 8 rows / block
    const dim3 GF((NFEATC + 7) / 8), GC((NCELLC + 7) / 8);

    // ---- stage 1: embeddings + input MLP ----
    embed_lrelu_kernel<<<GF, RWB, 0, stream>>>(P(11), feature_ids, f_ent, NFEATC);
    embed_lrelu_kernel<<<GC, RWB, 0, stream>>>(P(12), cell_ids, c_occ, NCELLC);
    linear_wmma_kernel<<<dim3(NCELLC / 16, 2), TB, 0, stream>>>(
        c_occ, P(13), nullptr, nullptr, P(14), acc_occ, NCELLC, HID, HID, 0);
    rownorm_kernel<<<GC, RWB, 0, stream>>>(acc_occ, hcell, P(17), P(18), NCELLC, 1, nullptr, 0);
    linear_wmma_kernel<<<dim3(NFEATC / 16, 2), TB, 0, stream>>>(
        f_ent, P(15), nullptr, nullptr, P(16), q, NFEATC, HID, HID, 0);
    rownorm_kernel<<<GF, RWB, 0, stream>>>(q, hfeat, P(19), P(20), NFEATC, 1, nullptr, 0);

    // ---- conv layers ----
    for (int l = 0; l < NLAYER; ++l) {
        const int cb = 21 + l * 9;                   // sage params base
        const int nb = 39 + l * 6;                   // conv_ln base (g,b per ln)
        const int ab = 51 + l * 2;                   // att base

        const size_t zn = 2 * NF128 + NC128 + 2 * NFEATC + NCELLC; // acc_* + cnt_* (contiguous)
        fill_kernel<<<(int)((zn + 255) / 256), 256, 0, stream>>>(acc_ent, 0.0f, zn);

        edge_scatter_kernel<<<(E_ENT + 7) / 8, RWB, 0, stream>>>(
            hcell, entail_src, entail_dst, entail_w, acc_ent, cnt_ent, E_ENT);
        edge_scatter_kernel<<<(E_PW + 7) / 8, RWB, 0, stream>>>(
            hfeat, pathway_src, pathway_dst, pathway_w, acc_pw, cnt_pw, E_PW);
        edge_scatter_kernel<<<(E_OCC + 7) / 8, RWB, 0, stream>>>(
            hfeat, occur_src, occur_dst, occur_w, acc_occ, cnt_occ, E_OCC);
        seg_div_kernel<<<GF, RWB, 0, stream>>>(acc_ent, cnt_ent, NFEATC);
        seg_div_kernel<<<GF, RWB, 0, stream>>>(acc_pw, cnt_pw, NFEATC);
        seg_div_kernel<<<GC, RWB, 0, stream>>>(acc_occ, cnt_occ, NCELLC);

        // sage linears: X_dst@W_self + neigh@W_neigh + b
        linear_wmma_kernel<<<dim3(NFEATC / 16, 2), TB, 0, stream>>>(
            hfeat, P(cb + 0), acc_ent, P(cb + 1), P(cb + 2), f_ent, NFEATC, HID, HID, 0);
        linear_wmma_kernel<<<dim3(NFEATC / 16, 2), TB, 0, stream>>>(
            hfeat, P(cb + 6), acc_pw, P(cb + 7), P(cb + 8), f_pw, NFEATC, HID, HID, 0);
        linear_wmma_kernel<<<dim3(NCELLC / 16, 2), TB, 0, stream>>>(
            hcell, P(cb + 3), acc_occ, P(cb + 4), P(cb + 5), c_occ, NCELLC, HID, HID, 0);
        // q = h0 @ attW + attb (before hfeat is overwritten)
        linear_wmma_kernel<<<dim3(NFEATC / 16, 2), TB, 0, stream>>>(
            hfeat, P(ab), nullptr, nullptr, P(ab + 1), q, NFEATC, HID, HID, 0);

        rownorm_kernel<<<GF, RWB, 0, stream>>>(f_ent, acc_ent, P(nb + 2), P(nb + 3), NFEATC, 0, nullptr, 0); // h1=LN(f_ent,ln1)
        rownorm_kernel<<<GF, RWB, 0, stream>>>(f_pw,  acc_pw,  P(nb + 4), P(nb + 5), NFEATC, 0, nullptr, 0); // h2=LN(f_pw ,ln2)
        attention_kernel<<<GF, RWB, 0, stream>>>(acc_ent, acc_pw, q, hfeat, NFEATC);
        // hcell = gelu(LN(c_occ, ln0)); also write hist[:, l*128:(l+1)*128]
        rownorm_kernel<<<GC, RWB, 0, stream>>>(c_occ, hcell, P(nb + 0), P(nb + 1), NCELLC, 2,
                                               hist + (size_t)l * HID, 2 * HID);
    }

    // ---- readout ----
    linear_wmma_kernel<<<dim3(NCELLC / 16, 4), TB, 0, stream>>>(
        hist, P(55), nullptr, nullptr, P(56), hread, NCELLC, 2 * HID, 2 * HID, 1);
    // pad 256x134 -> 256x144, GEMM into padded scratch, then slice to d_out
    pad_w_kernel<<<(2 * HID * NOUTP + 255) / 256, 256, 0, stream>>>(P(57), P(58), Wp, bp);
    linear_wmma_kernel<<<dim3(NCELLC / 16, NOUTP / (16 * LIN_WAVES) + 1), TB, 0, stream>>>(
        hread, Wp, nullptr, nullptr, bp, Yp, NCELLC, 2 * HID, NOUTP, 0);
    slice_out_kernel<<<(NCELLC * NOUTC + 255) / 256, 256, 0, stream>>>(Yp, (float*)d_out);
}